// tempoalNet_29360396435905
// MI455X (gfx1250) — compile-verified
//
#include <hip/hip_runtime.h>
#include <math.h>

#define NB 9
#define NCH 22
#define NT 1000
#define MM 48
#define CC 432
#define SFN 4
#define SEG 250
#define BATCH 256
#define NCLASS 4
#define NPAIR 253   // 22*23/2

typedef float v2f __attribute__((ext_vector_type(2)));
typedef float v8f __attribute__((ext_vector_type(8)));

// workspace layout (float offsets)
#define WS_ROWSUM 0      // 9*22 = 198
#define WS_GRAM   256    // 9*253 = 2277 -> ends 2533
#define WS_SS     2560   // 432*2 = 864  -> ends 3424
#define WS_XLV    3584   // 256*432*4 = 442368

// ---------------- zero the atomic-accumulated stats region ----------------
__global__ void k_zero(float* ws) {
    int i = blockIdx.x * blockDim.x + threadIdx.x;
    if (i < 2560) ws[i] = 0.f;
}

// ---------------- pass 1: per-band row sums + Gram matrices ----------------
__global__ __launch_bounds__(256) void k_stats(const float* __restrict__ x,
                                               float* __restrict__ ws) {
    const int b = blockIdx.x / NB;
    const int g = blockIdx.x % NB;
    const int tid = threadIdx.x;
    __shared__ float Xa[NCH][SEG + 6];

    int pi = 0, pj = 0;
    if (tid < NPAIR) {
        int rem = tid, i = 0;
        while (rem >= NCH - i) { rem -= NCH - i; ++i; }
        pi = i; pj = i + rem;
    }
    float accP = 0.f, accR = 0.f;
    const float* xb = x + ((size_t)(b * NB + g)) * NCH * NT;

    for (int seg = 0; seg < SFN; ++seg) {
        for (int idx = tid; idx < NCH * SEG; idx += 256) {
            int h = idx / SEG, t = idx % SEG;
            Xa[h][t] = xb[h * NT + seg * SEG + t];
        }
        __syncthreads();
        if (tid < NPAIR) {
            float a = 0.f;
            #pragma unroll 5
            for (int t = 0; t < SEG; ++t) a += Xa[pi][t] * Xa[pj][t];
            accP += a;
        }
        if (tid < NCH) {
            float r = 0.f;
            #pragma unroll 5
            for (int t = 0; t < SEG; ++t) r += Xa[tid][t];
            accR += r;
        }
        __syncthreads();
    }
    if (tid < NPAIR) atomicAdd(&ws[WS_GRAM + g * NPAIR + tid], accP);
    if (tid < NCH)   atomicAdd(&ws[WS_ROWSUM + g * NCH + tid], accR);
}

// ---------------- BN scale/shift per channel (conv bias folded) ----------------
__global__ __launch_bounds__(512) void k_bnparams(const float* __restrict__ conv_w,
                                                  const float* __restrict__ conv_b,
                                                  const float* __restrict__ gamma,
                                                  const float* __restrict__ beta,
                                                  float* __restrict__ ws) {
    int c = threadIdx.x;
    if (c >= CC) return;
    int g = c / MM, m = c % MM;
    float w[NCH];
    #pragma unroll
    for (int i = 0; i < NCH; ++i) w[i] = conv_w[(g * MM + m) * NCH + i];
    const float* rs = ws + WS_ROWSUM + g * NCH;
    const float* gr = ws + WS_GRAM + g * NPAIR;
    float sd = 0.f;
    #pragma unroll
    for (int i = 0; i < NCH; ++i) sd += w[i] * rs[i];
    float q = 0.f;
    int p = 0;
    for (int i = 0; i < NCH; ++i)
        for (int j = i; j < NCH; ++j, ++p) {
            float term = w[i] * w[j] * gr[p];
            q += (i == j) ? term : 2.f * term;
        }
    const float invN = 1.f / (float)(BATCH * NT);
    float bias = conv_b[c];
    float mean = sd * invN + bias;                     // E[h] incl. bias
    float e2 = q * invN + 2.f * bias * sd * invN + bias * bias;
    float var = e2 - mean * mean;                      // biased var (BN training)
    float s = gamma[c] * rsqrtf(var + 1e-5f);
    float t2 = s * (bias - mean) + beta[c];            // z = s*conv_noBias + t2
    ws[WS_SS + 2 * c]     = s;
    ws[WS_SS + 2 * c + 1] = t2;
}

// ---------------- pass 2: WMMA conv + BN + swish + segment log-variance ----------------
__global__ __launch_bounds__(128) void k_main(const float* __restrict__ x,
                                              const float* __restrict__ conv_w,
                                              float* __restrict__ ws) {
    const int b = blockIdx.x / NB;
    const int g = blockIdx.x % NB;
    const int tid  = threadIdx.x;
    const int lane = tid & 31;
    const int wave = tid >> 5;
    const int nrow  = lane & 15;
    const int khalf = (lane >> 4) << 1;  // K offset within group of 4: 0 or 2
    const int rbase = (lane >> 4) << 3;  // C/D row offset: 0 or 8

    __shared__ float Xs[24][260];
    __shared__ float sSum[MM][SFN];
    __shared__ float sSq[MM][SFN];
    __shared__ float sS[MM], sT[MM];

    if (tid < MM) {
        sS[tid] = ws[WS_SS + 2 * (g * MM + tid)];
        sT[tid] = ws[WS_SS + 2 * (g * MM + tid) + 1];
    }
    if (tid < MM * SFN) { sSum[tid >> 2][tid & 3] = 0.f; sSq[tid >> 2][tid & 3] = 0.f; }
    // zero the pad cells of Xs once: async copies only write valid cells
    for (int idx = tid; idx < 24 * 256; idx += 128) {
        int h = idx >> 8, tcol = idx & 255;
        if (h >= NCH || tcol >= SEG) Xs[h][tcol] = 0.f;
    }

    // A fragments: weights, ISA 16x4 f32 A layout, reused for every column tile
    v2f afrag[3][6];
    #pragma unroll
    for (int mt = 0; mt < 3; ++mt) {
        int m = mt * 16 + nrow;
        const float* wr = conv_w + (g * MM + m) * NCH;
        #pragma unroll
        for (int ks = 0; ks < 6; ++ks) {
            int k0 = ks * 4 + khalf;
            afrag[mt][ks].x = (k0     < NCH) ? wr[k0]     : 0.f;
            afrag[mt][ks].y = (k0 + 1 < NCH) ? wr[k0 + 1] : 0.f;
        }
    }

    const float* xb = x + ((size_t)(b * NB + g)) * NCH * NT;
    const unsigned xs_base = (unsigned)(size_t)&Xs[0][0];  // LDS byte offset (flat addr[31:0])

    for (int seg = 0; seg < SFN; ++seg) {
        // async-stage the 22x250 slab into LDS (b64 = 2 floats per op, 250 = 125 pairs)
        for (int idx = tid; idx < NCH * 125; idx += 128) {
            int h = idx / 125, p = idx % 125;
            unsigned lds_addr = xs_base + (unsigned)(h * 260 + 2 * p) * 4u;
            const float* gsrc = xb + h * NT + seg * SEG + 2 * p;
            asm volatile("global_load_async_to_lds_b64 %0, %1, off"
                         :: "v"(lds_addr), "v"(gsrc) : "memory");
        }
        asm volatile("s_wait_asynccnt 0x0" ::: "memory");
        __syncthreads();

        float aS[3][8], aQ[3][8];
        #pragma unroll
        for (int mt = 0; mt < 3; ++mt)
            #pragma unroll
            for (int i = 0; i < 8; ++i) { aS[mt][i] = 0.f; aQ[mt][i] = 0.f; }

        for (int tt = wave; tt < 16; tt += 4) {   // 4 iterations per wave, uniform
            v2f bfrag[6];
            #pragma unroll
            for (int ks = 0; ks < 6; ++ks) {
                int kb = ks * 4 + khalf;
                bfrag[ks].x = Xs[kb][tt * 16 + nrow];
                bfrag[ks].y = Xs[kb + 1][tt * 16 + nrow];
            }
            const bool tvalid = (tt * 16 + nrow) < SEG;
            #pragma unroll
            for (int mt = 0; mt < 3; ++mt) {
                v8f acc = {0.f, 0.f, 0.f, 0.f, 0.f, 0.f, 0.f, 0.f};
                #pragma unroll
                for (int ks = 0; ks < 6; ++ks) {
                    acc = __builtin_amdgcn_wmma_f32_16x16x4_f32(
                        false, afrag[mt][ks], false, bfrag[ks],
                        (short)0, acc, false, false);
                }
                #pragma unroll
                for (int i = 0; i < 8; ++i) {
                    int cl = mt * 16 + rbase + i;
                    float z = sS[cl] * acc[i] + sT[cl];
                    float zs = z * __builtin_amdgcn_rcpf(1.f + __expf(-z));  // swish
                    zs = tvalid ? zs : 0.f;
                    aS[mt][i] += zs;
                    aQ[mt][i] += zs * zs;
                }
            }
        }
        // one fire-and-forget LDS atomic flush per lane per accumulator
        #pragma unroll
        for (int mt = 0; mt < 3; ++mt)
            #pragma unroll
            for (int i = 0; i < 8; ++i) {
                int cl = mt * 16 + rbase + i;
                atomicAdd(&sSum[cl][seg], aS[mt][i]);
                atomicAdd(&sSq[cl][seg], aQ[mt][i]);
            }
        __syncthreads();
    }

    if (tid < MM * SFN) {
        int cl = tid >> 2, s = tid & 3;
        float su = sSum[cl][s], sq = sSq[cl][s];
        float mean = su * (1.f / SEG);
        float var = (sq - (float)SEG * mean * mean) * (1.f / (SEG - 1));  // unbiased
        var = fminf(fmaxf(var, 1e-6f), 1e6f);
        ws[WS_XLV + ((size_t)(b * CC + g * MM + cl)) * SFN + s] = logf(var);
    }
}

// ---------------- head: scalar attentions + linear + log_softmax ----------------
__global__ __launch_bounds__(512) void k_head(const float* __restrict__ ws,
    const float* __restrict__ wq, const float* __restrict__ wk, const float* __restrict__ wv,
    const float* __restrict__ bq, const float* __restrict__ bk, const float* __restrict__ bv,
    const float* __restrict__ wo, const float* __restrict__ bo,
    const float* __restrict__ lin_w, const float* __restrict__ lin_b,
    float* __restrict__ out) {
    const int b = blockIdx.x;
    const int tid = threadIdx.x;
    float p[NCLASS] = {0.f, 0.f, 0.f, 0.f};
    if (tid < CC) {
        const int c = tid;
        float xl[SFN], q[SFN], k[SFN], v[SFN], o[SFN];
        const float* xp = ws + WS_XLV + ((size_t)(b * CC + c)) * SFN;
        #pragma unroll
        for (int s = 0; s < SFN; ++s) xl[s] = xp[s];
        #pragma unroll
        for (int s = 0; s < SFN; ++s) {
            q[s] = xl[s] * wq[c] + bq[c];
            k[s] = xl[s] * wk[c] + bk[c];
            v[s] = xl[s] * wv[c] + bv[c];
        }
        #pragma unroll
        for (int s = 0; s < SFN; ++s) {
            float sc[SFN]; float mx = -1e30f;
            #pragma unroll
            for (int t = 0; t < SFN; ++t) { sc[t] = q[s] * k[t]; mx = fmaxf(mx, sc[t]); }
            float den = 0.f, acc = 0.f;
            #pragma unroll
            for (int t = 0; t < SFN; ++t) { float e = __expf(sc[t] - mx); den += e; acc += e * v[t]; }
            o[s] = (acc * __builtin_amdgcn_rcpf(den)) * wo[c] + bo[c];
        }
        #pragma unroll
        for (int j = 0; j < NCLASS; ++j) {
            float pj = 0.f;
            #pragma unroll
            for (int s = 0; s < SFN; ++s) pj += o[s] * lin_w[(size_t)j * (CC * SFN) + c * SFN + s];
            p[j] = pj;
        }
    }
    __shared__ float red[NCLASS][512];
    #pragma unroll
    for (int j = 0; j < NCLASS; ++j) red[j][tid] = p[j];
    __syncthreads();
    for (int sft = 256; sft > 0; sft >>= 1) {
        if (tid < sft) {
            #pragma unroll
            for (int j = 0; j < NCLASS; ++j) red[j][tid] += red[j][tid + sft];
        }
        __syncthreads();
    }
    if (tid == 0) {
        float lg[NCLASS], mx = -1e30f;
        #pragma unroll
        for (int j = 0; j < NCLASS; ++j) { lg[j] = red[j][0] + lin_b[j]; mx = fmaxf(mx, lg[j]); }
        float s = 0.f;
        #pragma unroll
        for (int j = 0; j < NCLASS; ++j) s += __expf(lg[j] - mx);
        float lse = mx + logf(s);
        #pragma unroll
        for (int j = 0; j < NCLASS; ++j) out[b * NCLASS + j] = lg[j] - lse;
    }
}

extern "C" void kernel_launch(void* const* d_in, const int* in_sizes, int n_in,
                              void* d_out, int out_size, void* d_ws, size_t ws_size,
                              hipStream_t stream) {
    (void)in_sizes; (void)n_in; (void)out_size; (void)ws_size;
    const float* x      = (const float*)d_in[0];
    const float* conv_w = (const float*)d_in[1];
    const float* conv_b = (const float*)d_in[2];
    const float* gamma  = (const float*)d_in[3];
    const float* beta   = (const float*)d_in[4];
    const float* wq     = (const float*)d_in[5];
    const float* wk     = (const float*)d_in[6];
    const float* wv     = (const float*)d_in[7];
    const float* bq     = (const float*)d_in[8];
    const float* bk     = (const float*)d_in[9];
    const float* bv     = (const float*)d_in[10];
    const float* wo     = (const float*)d_in[11];
    const float* bo     = (const float*)d_in[12];
    const float* lin_w  = (const float*)d_in[13];
    const float* lin_b  = (const float*)d_in[14];
    float* ws  = (float*)d_ws;
    float* out = (float*)d_out;

    k_zero<<<10, 256, 0, stream>>>(ws);
    k_stats<<<BATCH * NB, 256, 0, stream>>>(x, ws);
    k_bnparams<<<1, 512, 0, stream>>>(conv_w, conv_b, gamma, beta, ws);
    k_main<<<BATCH * NB, 128, 0, stream>>>(x, conv_w, ws);
    k_head<<<BATCH, 512, 0, stream>>>(ws, wq, wk, wv, bq, bk, bv, wo, bo, lin_w, lin_b, out);
}